// Explore_Decoder_61641370632882
// MI455X (gfx1250) — compile-verified
//
#include <hip/hip_runtime.h>
#include <hip/hip_bf16.h>

#define BB 1024
#define SS 100
#define HH 128
#define NN 50000
#define STILES 7          // ceil(100/16) -> 112 padded rows
#define SPAD (STILES*16)  // 112
#define ASTRIDE 132       // LDS row stride (words): 132 % 64 == 4 -> conflict-free frag reads

typedef float v2f __attribute__((ext_vector_type(2)));
typedef float v8f __attribute__((ext_vector_type(8)));

// ---------------------------------------------------------------------------
// Kernel 1: per-batch fused attention.
//   a = all_memory[b] @ U_w   (WMMA f32 16x16x4, A tile staged in LDS)
//   score[s] = tanh(a + l) . V_w + V_b ; masked softmax ; out_e = alpha^T A
//   feats[b] = [out_e , last_memory[b]]
// ---------------------------------------------------------------------------
__global__ __launch_bounds__(128)
void feats_kernel(const float* __restrict__ am,      // [B,S,H]
                  const float* __restrict__ lastm,   // [B,H]
                  const unsigned char* __restrict__ mask, // [B,S] bool
                  const float* __restrict__ U,       // [H,H]
                  const float* __restrict__ W,       // [H,H]
                  const float* __restrict__ Vw,      // [H]
                  const float* __restrict__ Vb,      // [1]
                  float* __restrict__ feats)         // [B,2H]
{
    __shared__ float A[SPAD * ASTRIDE];   // 112x132 words = 57.8 KB
    __shared__ float lvec[HH];
    __shared__ float scores[128];
    __shared__ float red[128];

    const int b    = blockIdx.x;
    const int t    = threadIdx.x;          // 0..127
    const int w    = t >> 5;               // wave 0..3
    const int lane = t & 31;
    const int g    = lane >> 4;            // half-wave (K group)
    const int ln   = lane & 15;

    // ---- stage A tile (all_memory[b], rows 0..99; zero-pad 100..111) ----
    const float4* amb4 = (const float4*)(am + (size_t)b * SS * HH);
    for (int i = t; i < SS * (HH / 4); i += 128) {
        int s  = i >> 5;            // 32 float4 per row
        int k4 = (i & 31) << 2;
        float4 v = amb4[i];
        *(float4*)&A[s * ASTRIDE + k4] = v;
    }
    for (int i = t; i < (SPAD - SS) * (HH / 4); i += 128) {
        int s  = SS + (i >> 5);
        int k4 = (i & 31) << 2;
        *(float4*)&A[s * ASTRIDE + k4] = float4{0.f, 0.f, 0.f, 0.f};
    }
    // ---- l = last_memory[b] @ W (thread t owns output column t) ----
    {
        const float* lb = lastm + (size_t)b * HH;
        float acc = 0.f;
        for (int k = 0; k < HH; ++k) acc += lb[k] * W[k * HH + t];
        lvec[t] = acc;
    }
    scores[t] = 0.f;
    __syncthreads();

    // ---- WMMA: A(112x128) @ U(128x128); wave owns 2 column tiles ----
    const v8f vzero = {0.f,0.f,0.f,0.f,0.f,0.f,0.f,0.f};
    for (int nt = 0; nt < 2; ++nt) {
        const int ncol = ((w << 1) + nt) * 16 + ln;   // 0..127
        v8f acc[STILES];
        #pragma unroll
        for (int st = 0; st < STILES; ++st) acc[st] = vzero;

        for (int kk = 0; kk < 32; ++kk) {
            const int k0 = (kk << 2) + (g << 1);      // lane's K pair
            v2f bf;
            bf.x = U[(size_t)k0 * HH + ncol];
            bf.y = U[(size_t)(k0 + 1) * HH + ncol];
            #pragma unroll
            for (int st = 0; st < STILES; ++st) {
                const int row = st * 16 + ln;
                v2f af;
                af.x = A[row * ASTRIDE + k0];
                af.y = A[row * ASTRIDE + k0 + 1];
                acc[st] = __builtin_amdgcn_wmma_f32_16x16x4_f32(
                    false, af, false, bf, (short)0, acc[st], false, false);
            }
        }
        // epilogue: tanh(a+l)*Vw, reduce over the 16 columns of this half-wave
        const float lval = lvec[ncol];
        const float vwv  = Vw[ncol];
        #pragma unroll
        for (int st = 0; st < STILES; ++st) {
            #pragma unroll
            for (int r = 0; r < 8; ++r) {
                float v = tanhf(acc[st][r] + lval) * vwv;
                v += __shfl_xor(v, 1, 32);
                v += __shfl_xor(v, 2, 32);
                v += __shfl_xor(v, 4, 32);
                v += __shfl_xor(v, 8, 32);
                if (ln == 0) {
                    int s = st * 16 + r + (g << 3);
                    atomicAdd(&scores[s], v);
                }
            }
        }
    }
    __syncthreads();

    // ---- masked softmax over s = 0..99 ----
    float sc;
    if (t < SS) {
        sc = scores[t] + Vb[0];
        if (mask[(size_t)b * SS + t]) sc = -1e9f;
    } else {
        sc = -INFINITY;
    }
    red[t] = sc;
    __syncthreads();
    for (int off = 64; off > 0; off >>= 1) {
        if (t < off) red[t] = fmaxf(red[t], red[t + off]);
        __syncthreads();
    }
    const float mx = red[0];
    __syncthreads();
    float e = (t < SS) ? expf(sc - mx) : 0.f;
    red[t] = e;
    __syncthreads();
    for (int off = 64; off > 0; off >>= 1) {
        if (t < off) red[t] += red[t + off];
        __syncthreads();
    }
    const float inv = 1.f / red[0];
    __syncthreads();
    scores[t] = e * inv;          // alpha (0 for s >= 100)
    __syncthreads();

    // ---- out_e[h] = sum_s alpha[s] * A[s][h]; feats = [out_e, last] ----
    float acc = 0.f;
    for (int s = 0; s < SS; ++s) acc += scores[s] * A[s * ASTRIDE + t];
    feats[(size_t)b * (2 * HH) + t]      = acc;
    feats[(size_t)b * (2 * HH) + HH + t] = lastm[(size_t)b * HH + t];
}

// ---------------------------------------------------------------------------
// Kernel 2: logits = feats[B,256] @ E_w[256,N], fused sigmoid.
// grid (ceil(N/128), B/64); 8 waves/block; each wave owns a 64(M)x16(N) strip
// (4 m-tiles register-blocked -> each E_w fragment feeds 4 WMMAs, E_w is
// re-read from L2 only 16x instead of 64x). K split into two 128-wide LDS
// phases so the 64-row feats tile fits padded (conflict-free) in 33.8 KB LDS.
// ---------------------------------------------------------------------------
#define K2_ROWS 64
#define K2_MT   4
#define K2_STRIDE 132     // 132 % 64 == 4 -> conflict-free frag reads

__global__ __launch_bounds__(256)
void logits_kernel(const float* __restrict__ feats,  // [B,256]
                   const float* __restrict__ Ew,     // [256,N]
                   float* __restrict__ out)          // [B,N]
{
    __shared__ float F[K2_ROWS * K2_STRIDE];   // 64x132 words = 33.8 KB

    const int t    = threadIdx.x;
    const int w    = t >> 5;
    const int lane = t & 31;
    const int g    = lane >> 4;
    const int ln   = lane & 15;
    const int m0   = blockIdx.y * K2_ROWS;
    const int ncol = blockIdx.x * 128 + w * 16 + ln;
    const int nc   = (ncol < NN) ? ncol : (NN - 1);  // clamped for loads

    const v8f vzero = {0.f,0.f,0.f,0.f,0.f,0.f,0.f,0.f};
    v8f acc[K2_MT];
    #pragma unroll
    for (int mt = 0; mt < K2_MT; ++mt) acc[mt] = vzero;

    for (int ph = 0; ph < 2; ++ph) {
        __syncthreads();   // protect F reads of previous phase
        // stage 64 x 128 feats sub-tile (K columns ph*128 .. ph*128+127)
        for (int i = t; i < K2_ROWS * 32; i += 256) {   // 2048 float4
            int r  = i >> 5;
            int k4 = (i & 31) << 2;
            float4 v = *(const float4*)&feats[(size_t)(m0 + r) * 256 + ph * 128 + k4];
            *(float4*)&F[r * K2_STRIDE + k4] = v;
        }
        __syncthreads();

        for (int kk = 0; kk < 32; ++kk) {
            const int k0 = (kk << 2) + (g << 1);        // lane's local K pair
            const int kg = ph * 128 + k0;               // global K row of Ew
            v2f bf;
            bf.x = Ew[(size_t)kg * NN + nc];
            bf.y = Ew[(size_t)(kg + 1) * NN + nc];
            if (kk < 30)   // pull upcoming B rows toward the WGP (global_prefetch_b8)
                __builtin_prefetch(&Ew[(size_t)(kg + 8) * NN + nc], 0, 1);
            #pragma unroll
            for (int mt = 0; mt < K2_MT; ++mt) {
                const int row = mt * 16 + ln;
                v2f af;
                af.x = F[row * K2_STRIDE + k0];
                af.y = F[row * K2_STRIDE + k0 + 1];
                acc[mt] = __builtin_amdgcn_wmma_f32_16x16x4_f32(
                    false, af, false, bf, (short)0, acc[mt], false, false);
            }
        }
    }

    if (ncol < NN) {
        #pragma unroll
        for (int mt = 0; mt < K2_MT; ++mt) {
            #pragma unroll
            for (int r = 0; r < 8; ++r) {
                int m = m0 + mt * 16 + r + (g << 3);
                float x = acc[mt][r];
                out[(size_t)m * NN + ncol] = 1.f / (1.f + expf(-x));
            }
        }
    }
}

// ---------------------------------------------------------------------------
// Kernel 3: seen-item mask. sigmoid(-inf) == 0, so just write 0.
// ---------------------------------------------------------------------------
__global__ __launch_bounds__(256)
void seen_kernel(const int* __restrict__ item_seq,  // [B,S]
                 float* __restrict__ out)           // [B,N]
{
    int idx = blockIdx.x * 256 + threadIdx.x;
    if (idx >= BB * SS) return;
    int b    = idx / SS;
    int item = item_seq[idx];
    if (item > 0) out[(size_t)b * NN + item] = 0.f;
}

extern "C" void kernel_launch(void* const* d_in, const int* in_sizes, int n_in,
                              void* d_out, int out_size, void* d_ws, size_t ws_size,
                              hipStream_t stream) {
    const float*         am    = (const float*)d_in[0];         // all_memory [B,S,H]
    const float*         lastm = (const float*)d_in[1];         // last_memory [B,H]
    const int*           iseq  = (const int*)d_in[2];           // item_seq [B,S]
    const unsigned char* mask  = (const unsigned char*)d_in[3]; // mask [B,S] bool
    const float*         U     = (const float*)d_in[4];         // U_w [H,H]
    const float*         W     = (const float*)d_in[5];         // W_w [H,H]
    const float*         Vw    = (const float*)d_in[6];         // V_w [H,1]
    const float*         Vb    = (const float*)d_in[7];         // V_b [1]
    const float*         Ew    = (const float*)d_in[8];         // E_w [2H,N]
    float*               out   = (float*)d_out;
    float*               feats = (float*)d_ws;                  // [B,2H] = 1 MB scratch

    feats_kernel<<<BB, 128, 0, stream>>>(am, lastm, mask, U, W, Vw, Vb, feats);

    dim3 grid2((NN + 127) / 128, BB / K2_ROWS);
    logits_kernel<<<grid2, 256, 0, stream>>>(feats, Ew, out);

    seen_kernel<<<(BB * SS + 255) / 256, 256, 0, stream>>>(iseq, out);
}